// WaveletPreprocessing_40054865002821
// MI455X (gfx1250) — compile-verified
//
#include <hip/hip_runtime.h>
#include <hip/hip_bf16.h>

// Haar DWT (level 1) + inverse == per-2x2-block butterfly. Fully fused,
// single pass: 1 GB traffic @ 23.3 TB/s => ~43 us roofline. Bandwidth-bound;
// use CDNA5 async global->LDS staging (ASYNCcnt) + NT b128 stores.

typedef float v4f __attribute__((ext_vector_type(4)));
typedef int v4i_vec __attribute__((vector_size(16)));
typedef __attribute__((address_space(1))) v4i_vec* gptr_v4i;  // global int4*
typedef __attribute__((address_space(3))) v4i_vec* lptr_v4i;  // LDS int4*

#define S_HAAR 0.7071067811865476f  // matches reference f32 rounding

// ---- CDNA5 async global->LDS path (guarded; fallback keeps compile green) --
#if defined(__has_builtin)
#  if __has_builtin(__builtin_amdgcn_global_load_async_to_lds_b128)
#    define HAVE_ASYNC_LDS 1
#  endif
#endif
#ifndef HAVE_ASYNC_LDS
#  define HAVE_ASYNC_LDS 0
#endif

#if HAVE_ASYNC_LDS
#  if defined(__has_builtin) && __has_builtin(__builtin_amdgcn_s_wait_asynccnt)
#    define WAIT_ASYNC(n) __builtin_amdgcn_s_wait_asynccnt(n)
#  else
#    define WAIT_ASYNC(n) asm volatile("s_wait_asynccnt %0" ::"i"(n) : "memory")
#  endif
#else
#  define WAIT_ASYNC(n) ((void)0)
#endif

// Stage one 8KB tile (4 rows x 512 f32) into LDS. Each of 256 lanes moves
// 32 bytes via two b128 async transfers (per-wave ASYNCcnt += 2).
__device__ __forceinline__ void stage_tile(const float* __restrict__ in,
                                           float* lds_base, long long tile,
                                           unsigned lane) {
  const float* gp = in + tile * 2048ll + (lane << 2);
  float* lp = lds_base + (lane << 2);
#if HAVE_ASYNC_LDS
  __builtin_amdgcn_global_load_async_to_lds_b128((gptr_v4i)gp, (lptr_v4i)lp, 0,
                                                 0);
  __builtin_amdgcn_global_load_async_to_lds_b128(
      (gptr_v4i)(gp + 1024), (lptr_v4i)(lp + 1024), 0, 0);
#else
  *(v4f*)lp = __builtin_nontemporal_load((const v4f*)gp);
  *(v4f*)(lp + 1024) = __builtin_nontemporal_load((const v4f*)(gp + 1024));
#endif
}

// Exact reference op order for one 2x2 block [a b; c d]:
// rows fwd -> cols fwd -> cols inv -> rows inv. No a*b+c patterns, so
// -ffp-contract cannot perturb rounding vs. the JAX reference.
struct Quad {
  float oa, ob, oc, od;  // oa=(2i,2j) ob=(2i,2j+1) oc=(2i+1,2j) od=(2i+1,2j+1)
};

__device__ __forceinline__ Quad haar2x2(float a, float b, float c, float d) {
  const float s = S_HAAR;
  float lo0 = (a + c) * s, lo1 = (b + d) * s;  // _dwt_rows
  float hi0 = (a - c) * s, hi1 = (b - d) * s;
  float ll = (lo0 + lo1) * s, lh = (lo0 - lo1) * s;  // _dwt_cols
  float hl = (hi0 + hi1) * s, hh = (hi0 - hi1) * s;
  float lr0 = (ll + lh) * s, lr1 = (ll - lh) * s;  // _idwt_cols
  float hr0 = (hl + hh) * s, hr1 = (hl - hh) * s;
  Quad q;
  q.oa = (lr0 + hr0) * s;  // _idwt_rows: even row
  q.ob = (lr1 + hr1) * s;
  q.oc = (lr0 - hr0) * s;  // odd row
  q.od = (lr1 - hr1) * s;
  return q;
}

__global__ __launch_bounds__(256) void haar_roundtrip_kernel(
    const float* __restrict__ in, float* __restrict__ out, int ntiles) {
  __shared__ __align__(16) float lds[2][2048];  // double-buffered 8KB tiles
  const unsigned lane = threadIdx.x;            // 0..255
  const int stride = gridDim.x;
  const int t0 = blockIdx.x;
  if (t0 >= ntiles) return;

  stage_tile(in, &lds[0][0], (long long)t0, lane);

  int buf = 0;
  for (int t = t0; t < ntiles; t += stride, buf ^= 1) {
    const int tn = t + stride;
    if (tn < ntiles) {
      stage_tile(in, &lds[buf ^ 1][0], (long long)tn, lane);
      WAIT_ASYNC(2);  // current tile's 2 per-wave async loads done; next 2 fly
    } else {
      WAIT_ASYNC(0);
    }
    __syncthreads();  // cross-wave visibility of the staged tile

    // Tile = rows {0,1,2,3} x 512 cols. Lane -> row-pair p, col group c.
    const int p = (lane >> 7) & 1;
    const int c = (int)(lane & 127u) << 2;
    const float* r0p = &lds[buf][p * 1024 + c];        // even row of pair
    const float* r1p = &lds[buf][p * 1024 + 512 + c];  // odd row of pair
    v4f r0 = *(const v4f*)r0p;
    v4f r1 = *(const v4f*)r1p;

    Quad q0 = haar2x2(r0.x, r0.y, r1.x, r1.y);
    Quad q1 = haar2x2(r0.z, r0.w, r1.z, r1.w);
    v4f o0 = {q0.oa, q0.ob, q1.oa, q1.ob};  // even row of pair
    v4f o1 = {q0.oc, q0.od, q1.oc, q1.od};  // odd row of pair

    float* ob = out + (long long)t * 2048ll;
    __builtin_nontemporal_store(o0, (v4f*)(ob + p * 1024 + c));
    __builtin_nontemporal_store(o1, (v4f*)(ob + p * 1024 + 512 + c));

    __syncthreads();  // tile buffer reused by next iteration's async stage
  }
}

extern "C" void kernel_launch(void* const* d_in, const int* in_sizes, int n_in,
                              void* d_out, int out_size, void* d_ws,
                              size_t ws_size, hipStream_t stream) {
  (void)n_in; (void)d_ws; (void)ws_size;
  const float* x = (const float*)d_in[0];
  float* out = (float*)d_out;

  // 16*32*512*512 = 134,217,728 f32; tile = 2048 f32 (4 rows of 512).
  long long n = (long long)out_size;
  int ntiles = (int)(n / 2048ll);
  if (ntiles <= 0) return;
  int grid = ntiles < 8192 ? ntiles : 8192;
  haar_roundtrip_kernel<<<grid, 256, 0, stream>>>(x, out, ntiles);
}